// Fcaf3DNeckWithHead_my_67602785239556
// MI455X (gfx1250) — compile-verified
//
#include <hip/hip_runtime.h>
#include <hip/hip_bf16.h>
#include <stdint.h>

// ---------------------------------------------------------------------------
// FCAF3D head proposal selection for MI455X (gfx1250, wave32).
// Memory-bound selection (~70 MB moved -> ~3us @ 23.3 TB/s). No matrix FLOPs
// -> no WMMA. CDNA5 path used: double-buffered async global->LDS staging
// (global_load_async_to_lds_b128 + in-order ASYNCcnt pipelining via
// s_wait_asynccnt) for the 72B-stride cls_scores stream, plus b128 gathers.
// ---------------------------------------------------------------------------

#define BATCH   8
#define NPTS    100000
#define NCLS    18
#define NBINS   8192          // float bits >> 17 (scores in (0,1) -> bin < 8128)
#define TILE    192           // points staged per buffer (13824 B)
#define TILE_BYTES (TILE * NCLS * 4)
#define LPT     4             // b128 issues per thread per tile (864/256 -> 4)
#define CAND_MAX 4096
#define NCROSS  1024
#define TOPK    256

__device__ __forceinline__ void async_ld_b128(uint32_t lds_addr, uint64_t gaddr) {
  asm volatile("global_load_async_to_lds_b128 %0, %1, off"
               :: "v"(lds_addr), "v"(gaddr) : "memory");
}
__device__ __forceinline__ void wait_async0() {
  asm volatile("s_wait_asynccnt 0" ::: "memory");
}
__device__ __forceinline__ void wait_async_le4() {
  // async loads complete in order: <=4 outstanding  => previous tile resident
  asm volatile("s_wait_asynccnt 0x4" ::: "memory");
}

__device__ __forceinline__ float fast_sigmoid(float x) {
  return __builtin_amdgcn_rcpf(1.0f + __expf(-x));   // monotone; ordering-safe
}

// ---------------------------------------------------------------------------
// K1: fused max-over-class + sigmoid*sigmoid score + per-row histogram.
// grid = (128, B), block = 256 (8 waves).
// LDS: 32KB hist + 2 x 13.5KB tile buffers = 59KB.
// ---------------------------------------------------------------------------
__global__ void k_score_hist(const float* __restrict__ cen,
                             const float* __restrict__ cls,
                             float* __restrict__ maxs,
                             unsigned* __restrict__ ghist) {
  __shared__ unsigned s_hist[NBINS];
  __shared__ float s_tile[2][TILE * NCLS];

  const int b = blockIdx.y;
  for (int i = threadIdx.x; i < NBINS; i += blockDim.x) s_hist[i] = 0;

  const int ntiles = (NPTS + TILE - 1) / TILE;
  const int stride = gridDim.x;
  const uint64_t rowbase = (uint64_t)(uintptr_t)(cls + (size_t)b * NPTS * NCLS);
  const uint32_t lds0 = (uint32_t)(uintptr_t)(&s_tile[0][0]);
  const uint32_t lds1 = (uint32_t)(uintptr_t)(&s_tile[1][0]);

  // stage one tile: 864 b128 / 256 threads -> 4 issues each; OOB lanes clamp
  // src & dst to the same safe offset (duplicate identical writes: benign).
  auto issue = [&](int tile, uint32_t lbase) {
    const int i0 = tile * TILE;
    int vp = NPTS - i0; if (vp > TILE) vp = TILE;
    const uint32_t clampOff = (uint32_t)((vp * NCLS * 4 - 16) & ~15);
    const uint64_t gbase = rowbase + (uint64_t)i0 * (NCLS * 4);
#pragma unroll
    for (int k = 0; k < LPT; ++k) {
      uint32_t off = (uint32_t)(threadIdx.x + k * 256) * 16u;
      if (off > clampOff) off = clampOff;
      async_ld_b128(lbase + off, gbase + off);
    }
  };

  int tile = blockIdx.x;
  int cur = 0;
  if (tile < ntiles) issue(tile, lds0);          // prologue: tile 0 -> buf 0

  for (; tile < ntiles; tile += stride) {
    const int nxt = tile + stride;               // uniform across block
    if (nxt < ntiles) {
      issue(nxt, cur ? lds0 : lds1);             // prefetch into other buffer
      wait_async_le4();                          // drain current tile's 4
    } else {
      wait_async0();
    }
    __syncthreads();                             // all waves' loads visible

    const int i0 = tile * TILE;
    int vp = NPTS - i0; if (vp > TILE) vp = TILE;
    const int t = threadIdx.x;
    if (t < vp) {
      const float* p = &s_tile[cur][t * NCLS];   // stride 18 over 64 banks: conflict-free
      float m = p[0];
#pragma unroll
      for (int c = 1; c < NCLS; ++c) m = fmaxf(m, p[c]);
      const int idx = i0 + t;
      const float sc = fast_sigmoid(m) * fast_sigmoid(cen[(size_t)b * NPTS + idx]);
      maxs[(size_t)b * NPTS + idx] = sc;
      atomicAdd(&s_hist[__float_as_uint(sc) >> 17], 1u);
    }
    __syncthreads();       // buffer 'cur' free for overwrite next iteration
    cur ^= 1;
  }

  for (int i = threadIdx.x; i < NBINS; i += blockDim.x) {
    const unsigned c = s_hist[i];
    if (c) atomicAdd(&ghist[(size_t)b * NBINS + i], c);
  }
}

// ---------------------------------------------------------------------------
// K2: per-row pivot bin = largest bin with cumulative-from-top >= 1024.
// grid = B, block = 256 (32 bins/thread), suffix-scan in LDS.
// ---------------------------------------------------------------------------
__global__ void k_pivot(const unsigned* __restrict__ ghist, int* __restrict__ pivot) {
  __shared__ unsigned s_part[256];
  const int b = blockIdx.x, t = threadIdx.x;
  const int base = t * 32;
  unsigned vals[32];
  unsigned own = 0;
#pragma unroll
  for (int i = 0; i < 32; ++i) { vals[i] = ghist[(size_t)b * NBINS + base + i]; own += vals[i]; }
  s_part[t] = own;
  __syncthreads();
  for (int off = 1; off < 256; off <<= 1) {   // inclusive suffix sum
    const unsigned o = (t + off < 256) ? s_part[t + off] : 0u;
    __syncthreads();
    s_part[t] += o;
    __syncthreads();
  }
  const unsigned incl = s_part[t];
  const unsigned above = incl - own;
  const unsigned K = NCROSS;
  if (above < K && incl >= K) {               // unique crossing thread
    unsigned cum = above;
    for (int i = 31; i >= 0; --i) {
      cum += vals[i];
      if (cum >= K) { pivot[b] = base + i; break; }
    }
  }
}

// ---------------------------------------------------------------------------
// K3: compact candidates (bin >= pivot) as u64 key: (scorebits<<32) | ~idx
// so descending sort gives jax's smaller-index-first tie-break.
// ---------------------------------------------------------------------------
__global__ void k_compact(const float* __restrict__ maxs,
                          const int* __restrict__ pivot,
                          unsigned* __restrict__ cand_cnt,
                          unsigned long long* __restrict__ cand) {
  const int b = blockIdx.y;
  const int pb = pivot[b];
  for (int i = blockIdx.x * blockDim.x + threadIdx.x; i < NPTS;
       i += gridDim.x * blockDim.x) {
    const unsigned bits = __float_as_uint(maxs[(size_t)b * NPTS + i]);
    if ((int)(bits >> 17) >= pb) {
      const unsigned pos = atomicAdd(&cand_cnt[b], 1u);
      if (pos < CAND_MAX)
        cand[(size_t)b * CAND_MAX + pos] =
            ((unsigned long long)bits << 32) | (unsigned)(~(unsigned)i);
    }
  }
}

// ---------------------------------------------------------------------------
// K4: per-row final selection. 1024 threads, 4096-key bitonic sort descending
// in LDS. Emits cross_inds (ws), sort_inds (out, as float + ws int), pts_sel.
// ---------------------------------------------------------------------------
__global__ void __launch_bounds__(1024)
k_select(const unsigned long long* __restrict__ cand,
         const unsigned* __restrict__ cand_cnt,
         const float* __restrict__ points,
         int* __restrict__ cross_inds,
         float* __restrict__ out_pts,
         float* __restrict__ out_sortinds,
         int* __restrict__ sel_inds) {
  __shared__ unsigned long long s_key[CAND_MAX];
  __shared__ int s_idx[TOPK];
  const int b = blockIdx.x, t = threadIdx.x;

  unsigned cnt = cand_cnt[b]; if (cnt > CAND_MAX) cnt = CAND_MAX;
  for (int i = t; i < CAND_MAX; i += 1024)
    s_key[i] = (i < (int)cnt) ? cand[(size_t)b * CAND_MAX + i] : 0ull;
  __syncthreads();

  // bitonic sort, descending
  for (int k = 2; k <= CAND_MAX; k <<= 1)
    for (int j = k >> 1; j > 0; j >>= 1) {
      for (int i = t; i < CAND_MAX; i += 1024) {
        const int p = i ^ j;
        if (p > i) {
          const unsigned long long a = s_key[i], c = s_key[p];
          const bool sw = ((i & k) == 0) ? (a < c) : (a > c);
          if (sw) { s_key[i] = c; s_key[p] = a; }
        }
      }
      __syncthreads();
    }

  // top-1024 -> cross attention indices (descending-score order, as in topk)
  cross_inds[b * NCROSS + t] = (int)(~(unsigned)(s_key[t] & 0xFFFFFFFFull));
  if (t < TOPK) s_idx[t] = (int)(~(unsigned)(s_key[t] & 0xFFFFFFFFull));
  __syncthreads();

  // sort the 256 proposal indices ascending (jnp.sort(inds))
  for (int k = 2; k <= TOPK; k <<= 1)
    for (int j = k >> 1; j > 0; j >>= 1) {
      if (t < TOPK) {
        const int i = t, p = i ^ j;
        if (p > i) {
          const int a = s_idx[i], c = s_idx[p];
          const bool sw = ((i & k) == 0) ? (a > c) : (a < c);
          if (sw) { s_idx[i] = c; s_idx[p] = a; }
        }
      }
      __syncthreads();
    }

  if (t < TOPK) {
    const int si = s_idx[t];
    sel_inds[b * TOPK + t] = si;
    out_sortinds[b * TOPK + t] = (float)si;   // d_out is float-typed
  }
  if (t < TOPK * 3) {                          // pts_sel gather (tiny)
    const int r = t / 3, comp = t % 3;
    out_pts[(size_t)b * TOPK * 3 + t] =
        points[((size_t)b * NPTS + s_idx[r]) * 3 + comp];
  }
}

// ---------------------------------------------------------------------------
// K5: wave-per-row feature gather: 32 lanes x float4 = 512B row (b128 loads).
// ---------------------------------------------------------------------------
__global__ void k_gather_feat(const float* __restrict__ feat,
                              const int* __restrict__ inds,
                              float* __restrict__ out,
                              int rows_per_b) {
  const int g = blockIdx.x * (blockDim.x >> 5) + (threadIdx.x >> 5);
  const int lane = threadIdx.x & 31;
  if (g >= BATCH * rows_per_b) return;
  const int b = g / rows_per_b;
  const int idx = inds[g];
  const float4* src = (const float4*)(feat + ((size_t)b * NPTS + idx) * 128);
  float4* dst = (float4*)(out + (size_t)g * 128);
  dst[lane] = src[lane];
}

// ---------------------------------------------------------------------------
extern "C" void kernel_launch(void* const* d_in, const int* in_sizes, int n_in,
                              void* d_out, int out_size, void* d_ws, size_t ws_size,
                              hipStream_t stream) {
  const float* cen  = (const float*)d_in[0];   // [8,100000]
  const float* cls  = (const float*)d_in[1];   // [8,100000,18]
  const float* pts  = (const float*)d_in[2];   // [8,100000,3]
  const float* feat = (const float*)d_in[3];   // [8,100000,128]
  float* out = (float*)d_out;

  // output layout (flat, return order)
  float* out_pts      = out;                               //  6144
  float* out_featsel  = out + 6144;                        //  262144
  float* out_sortinds = out + 6144 + 262144;               //  2048
  float* out_cross    = out + 6144 + 262144 + 2048;        //  1048576

  // workspace layout (16B-aligned chunks)
  char* ws = (char*)d_ws;
  float*              maxs     = (float*)(ws);                           // 3,200,000 B
  unsigned*           ghist    = (unsigned*)(ws + 3200000);              //   262,144 B
  unsigned*           cand_cnt = (unsigned*)(ws + 3200000 + 262144);     //       128 B
  int*                pivot    = (int*)(ws + 3200000 + 262144 + 128);    //       128 B
  unsigned long long* cand     = (unsigned long long*)(ws + 3200000 + 262144 + 256); // 262,144 B
  int*                crossids = (int*)(ws + 3200000 + 262144 + 256 + 262144);       //  32,768 B
  int*                selids   = (int*)(ws + 3200000 + 262144 + 256 + 262144 + 32768);

  // zero histogram + candidate counters (graph-capturable memset node)
  hipMemsetAsync(ghist, 0, 262144 + 128, stream);

  dim3 gScan(128, BATCH);
  k_score_hist<<<gScan, 256, 0, stream>>>(cen, cls, maxs, ghist);
  k_pivot<<<BATCH, 256, 0, stream>>>(ghist, pivot);
  k_compact<<<gScan, 256, 0, stream>>>(maxs, pivot, cand_cnt, cand);
  k_select<<<BATCH, 1024, 0, stream>>>(cand, cand_cnt, pts, crossids,
                                       out_pts, out_sortinds, selids);
  k_gather_feat<<<256, 256, 0, stream>>>(feat, selids, out_featsel, TOPK);     // 2048 waves
  k_gather_feat<<<1024, 256, 0, stream>>>(feat, crossids, out_cross, NCROSS);  // 8192 waves
}